// Attn_head_35983236006387
// MI455X (gfx1250) — compile-verified
//
#include <hip/hip_runtime.h>

// ---------------------------------------------------------------------------
// GAT attention head for MI455X (gfx1250, wave32).
// GEMMs use V_WMMA_F32_16X16X4_F32 (full fp32 — problem is bandwidth bound,
// dominated by the edge gather/scatter, so no precision tradeoff is needed).
// ---------------------------------------------------------------------------

typedef float v2f __attribute__((ext_vector_type(2)));
typedef float v8f __attribute__((ext_vector_type(8)));

#define N_NODES 50000
#define E_EDGES 800000
#define CIN     256
#define COUT    128
#define LDS_STRIDE 260   // 256 + 4 pad -> conflict-free column access (64 banks)

// ---------------------------------------------------------------------------
// Fused GEMM: [N,256] x [256,256] -> seq_fts [N,128] (ws) and
// residual + b_res + bias [N,128] written directly to d_out (scatter adds on top).
// Block: 256 threads = 8 waves, 16 output rows per block.
// Wave w computes column tile [16w,16w+16) of BOTH halves.
// ---------------------------------------------------------------------------
__global__ __launch_bounds__(256) void gemm_kernel(
    const float* __restrict__ seq,
    const float* __restrict__ Wseq,   // [128,256] row-major (out, in)
    const float* __restrict__ Wres,   // [128,256]
    const float* __restrict__ bias,   // [128]
    const float* __restrict__ bres,   // [128]
    float* __restrict__ seq_fts,      // [N,128]
    float* __restrict__ out)          // [N,128]
{
    __shared__ float As[16 * LDS_STRIDE];

    const int tid     = threadIdx.x;
    const int rowBase = blockIdx.x * 16;

    // Cooperative, coalesced load of the 16x256 A tile into LDS.
    for (int i = tid; i < 16 * CIN; i += 256) {
        const int r = i >> 8;
        const int c = i & 255;
        As[r * LDS_STRIDE + c] = seq[(size_t)(rowBase + r) * CIN + c];
    }
    __syncthreads();

    const int lane = tid & 31;
    const int wave = tid >> 5;
    const int half = lane >> 4;   // 0: K+{0,1}, 1: K+{2,3} (f32 A/B fragment layout)
    const int m    = lane & 15;   // row (A) / col (B) within tile
    const int col  = 16 * wave + m;               // 0..127 within each half
    const float* pB0 = Wseq + (size_t)col * CIN;  // B[k][n] = W[n][k]
    const float* pB1 = Wres + (size_t)col * CIN;

    v8f acc0 = {};
    v8f acc1 = {};

#pragma unroll 8
    for (int k0 = 0; k0 < CIN; k0 += 4) {
        const int kk = k0 + 2 * half;
        const float2 af = *(const float2*)(&As[m * LDS_STRIDE + kk]);
        const float2 b0 = *(const float2*)(pB0 + kk);
        const float2 b1 = *(const float2*)(pB1 + kk);
        v2f a  = { af.x, af.y };
        v2f vb0 = { b0.x, b0.y };
        v2f vb1 = { b1.x, b1.y };
        acc0 = __builtin_amdgcn_wmma_f32_16x16x4_f32(false, a, false, vb0,
                                                     (short)0, acc0, false, false);
        acc1 = __builtin_amdgcn_wmma_f32_16x16x4_f32(false, a, false, vb1,
                                                     (short)0, acc1, false, false);
    }

    // D layout: VGPR v <-> row M = v + 8*half, col = lane&15 within tile.
    const float badd = bres[col] + bias[col];
#pragma unroll
    for (int v = 0; v < 8; ++v) {
        const int row = rowBase + v + 8 * half;
        seq_fts[(size_t)row * COUT + col] = acc0[v];
        out[(size_t)row * COUT + col]     = acc1[v] + badd;
    }
}

// ---------------------------------------------------------------------------
// Per-node attention scalars: f1 = seq_fts . w_f1 + b_f1 (same for f2).
// One wave per node; 4 channels per lane; shuffle reduction.
// ---------------------------------------------------------------------------
__global__ __launch_bounds__(256) void f12_kernel(
    const float* __restrict__ seq_fts,
    const float* __restrict__ wf1, const float* __restrict__ bf1,
    const float* __restrict__ wf2, const float* __restrict__ bf2,
    float* __restrict__ f1, float* __restrict__ f2)
{
    const int node = blockIdx.x * 8 + (threadIdx.x >> 5);
    if (node >= N_NODES) return;
    const int lane = threadIdx.x & 31;

    const float4 x = *(const float4*)(seq_fts + (size_t)node * COUT + lane * 4);
    const float4 a = *(const float4*)(wf1 + lane * 4);
    const float4 b = *(const float4*)(wf2 + lane * 4);

    float s1 = x.x * a.x + x.y * a.y + x.z * a.z + x.w * a.w;
    float s2 = x.x * b.x + x.y * b.y + x.z * b.z + x.w * b.w;
    for (int off = 16; off; off >>= 1) {
        s1 += __shfl_down(s1, off, 32);
        s2 += __shfl_down(s2, off, 32);
    }
    if (lane == 0) {
        f1[node] = s1 + bf1[0];
        f2[node] = s2 + bf2[0];
    }
}

// ---------------------------------------------------------------------------
// Global softmax over edges, pass 1: e = selu(f1[src] + f2[dst]); track max.
// ---------------------------------------------------------------------------
__device__ inline float atomicMaxF(float* addr, float v) {
    return (v >= 0.0f)
        ? __int_as_float(atomicMax((int*)addr, __float_as_int(v)))
        : __uint_as_float(atomicMin((unsigned int*)addr, __float_as_uint(v)));
}

__global__ __launch_bounds__(256) void logits_kernel(
    const long long* __restrict__ ei,     // [2,E] int64
    const float* __restrict__ f1, const float* __restrict__ f2,
    float* __restrict__ e, float* __restrict__ gmax)
{
    const int i = blockIdx.x * blockDim.x + threadIdx.x;
    float x = -__builtin_inff();
    if (i < E_EDGES) {
        const int s = (int)ei[i];
        const int d = (int)ei[E_EDGES + i];
        const float z = f1[s] + f2[d];
        const float scale = 1.0507009873554805f;
        const float alpha = 1.6732632423543772f;
        x = scale * (z > 0.0f ? z : alpha * (__expf(z) - 1.0f));
        e[i] = x;
    }
    for (int off = 16; off; off >>= 1) x = fmaxf(x, __shfl_down(x, off, 32));
    if ((threadIdx.x & 31) == 0) atomicMaxF(gmax, x);
}

// ---------------------------------------------------------------------------
// Softmax pass 2: p = exp(e - max) stored in place; accumulate global sum.
// ---------------------------------------------------------------------------
__global__ __launch_bounds__(256) void expsum_kernel(
    float* __restrict__ e, const float* __restrict__ gmax,
    float* __restrict__ gsum)
{
    const int i = blockIdx.x * blockDim.x + threadIdx.x;
    const float mx = gmax[0];
    float p = 0.0f;
    if (i < E_EDGES) {
        p = __expf(e[i] - mx);
        e[i] = p;
    }
    for (int off = 16; off; off >>= 1) p += __shfl_down(p, off, 32);
    if ((threadIdx.x & 31) == 0) atomicAdd(gsum, p);
}

// ---------------------------------------------------------------------------
// Edge scatter: w = (p/sum) * seq_fts[dst]; out[src] += w; out[dst] += w.
// One wave per edge, 4 channels per lane (float4 gather, 8 atomics/lane).
// ---------------------------------------------------------------------------
__global__ __launch_bounds__(256) void scatter_kernel(
    const long long* __restrict__ ei,
    const float* __restrict__ p, const float* __restrict__ gsum,
    const float* __restrict__ seq_fts, float* __restrict__ out)
{
    const int edge = blockIdx.x * 8 + (threadIdx.x >> 5);
    if (edge >= E_EDGES) return;
    const int lane = threadIdx.x & 31;

    const int s = (int)ei[edge];
    const int d = (int)ei[E_EDGES + edge];
    const float coef = p[edge] / gsum[0];

    const float4 x = *(const float4*)(seq_fts + (size_t)d * COUT + lane * 4);
    const float wx = coef * x.x, wy = coef * x.y, wz = coef * x.z, ww = coef * x.w;

    float* ps = out + (size_t)s * COUT + lane * 4;
    atomicAdd(ps + 0, wx); atomicAdd(ps + 1, wy);
    atomicAdd(ps + 2, wz); atomicAdd(ps + 3, ww);

    float* pd = out + (size_t)d * COUT + lane * 4;
    atomicAdd(pd + 0, wx); atomicAdd(pd + 1, wy);
    atomicAdd(pd + 2, wz); atomicAdd(pd + 3, ww);
}

// ---------------------------------------------------------------------------
// Final ELU (alpha=1) in place over d_out.
// ---------------------------------------------------------------------------
__global__ __launch_bounds__(256) void elu_kernel(float* __restrict__ out)
{
    const int i = blockIdx.x * blockDim.x + threadIdx.x;
    if (i < (N_NODES * COUT) / 4) {
        float4 x = ((float4*)out)[i];
        x.x = x.x > 0.0f ? x.x : (__expf(x.x) - 1.0f);
        x.y = x.y > 0.0f ? x.y : (__expf(x.y) - 1.0f);
        x.z = x.z > 0.0f ? x.z : (__expf(x.z) - 1.0f);
        x.w = x.w > 0.0f ? x.w : (__expf(x.w) - 1.0f);
        ((float4*)out)[i] = x;
    }
}

__global__ void init_kernel(float* __restrict__ scalars)
{
    scalars[0] = -__builtin_inff(); // running max
    scalars[1] = 0.0f;              // running sum of exp
}

// ---------------------------------------------------------------------------
extern "C" void kernel_launch(void* const* d_in, const int* in_sizes, int n_in,
                              void* d_out, int out_size, void* d_ws, size_t ws_size,
                              hipStream_t stream)
{
    const float*     seq  = (const float*)d_in[0];
    const long long* ei   = (const long long*)d_in[1];
    const float*     Wseq = (const float*)d_in[2];
    const float*     wf1  = (const float*)d_in[3];
    const float*     bf1  = (const float*)d_in[4];
    const float*     wf2  = (const float*)d_in[5];
    const float*     bf2  = (const float*)d_in[6];
    const float*     bias = (const float*)d_in[7];
    const float*     Wres = (const float*)d_in[8];
    const float*     bres = (const float*)d_in[9];
    float* out = (float*)d_out;

    // Workspace layout (floats): seq_fts[N*128] | f1[N] | f2[N] | p[E] | scalars[2]
    float* ws      = (float*)d_ws;
    float* seq_fts = ws;
    float* f1      = seq_fts + (size_t)N_NODES * COUT;
    float* f2      = f1 + N_NODES;
    float* pe      = f2 + N_NODES;
    float* scal    = pe + E_EDGES;

    init_kernel<<<1, 32, 0, stream>>>(scal);

    // N_NODES = 50000 = 3125 * 16, exact tiling.
    gemm_kernel<<<N_NODES / 16, 256, 0, stream>>>(seq, Wseq, Wres, bias, bres,
                                                  seq_fts, out);

    f12_kernel<<<(N_NODES + 7) / 8, 256, 0, stream>>>(seq_fts, wf1, bf1, wf2, bf2,
                                                      f1, f2);

    logits_kernel<<<(E_EDGES + 255) / 256, 256, 0, stream>>>(ei, f1, f2, pe, scal);

    expsum_kernel<<<(E_EDGES + 255) / 256, 256, 0, stream>>>(pe, scal, scal + 1);

    scatter_kernel<<<(E_EDGES + 7) / 8, 256, 0, stream>>>(ei, pe, scal + 1,
                                                          seq_fts, out);

    elu_kernel<<<((N_NODES * COUT / 4) + 255) / 256, 256, 0, stream>>>(out);
}